// VirtualBuffer_9354438771004
// MI455X (gfx1250) — compile-verified
//
#include <hip/hip_runtime.h>
#include <cstdint>

// Problem constants (match the reference: B=2048, O=128, D=1024, K=8).
#define O_BANKS        128
#define D_DIM          1024
#define K_SEL          8
#define ROWS_PER_BLOCK 4
#define THREADS        256   // D_DIM/4 float4 lanes per row

typedef float v4f __attribute__((ext_vector_type(4)));
typedef int   v4i __attribute__((vector_size(16)));   // matches builtin's param

// ---------------------------------------------------------------------------
// CDNA5 async global->LDS copy (GLOBAL_LOAD_ASYNC_TO_LDS_B128, ASYNCcnt).
// Builtin signature (from hipcc diagnostic): param 0 is
//   'int __attribute__((vector_size(16))) __device__ *'  (addrspace(1) v4i*)
// so cast to addrspace-qualified v4i pointers explicitly.
// ---------------------------------------------------------------------------
#if defined(__has_builtin) && __has_builtin(__builtin_amdgcn_global_load_async_to_lds_b128)
#define HAVE_ASYNC_LDS_BUILTIN 1
typedef __attribute__((address_space(1))) v4i* g4p_t;   // global
typedef __attribute__((address_space(3))) v4i* l4p_t;   // LDS
#endif

__device__ __forceinline__ void async_load_f4_to_lds(const float* gsrc,
                                                     float* lds_dst) {
#if defined(HAVE_ASYNC_LDS_BUILTIN)
  __builtin_amdgcn_global_load_async_to_lds_b128(
      (g4p_t)gsrc, (l4p_t)lds_dst, /*offset=*/0, /*cpol=*/0);
#else
  // Low 32 bits of a generic pointer into the LDS aperture are the LDS byte
  // address (CDNA5 ISA).
  uint32_t lds_addr = (uint32_t)(uintptr_t)lds_dst;
  uint64_t gaddr    = (uint64_t)(uintptr_t)gsrc;
  asm volatile("global_load_async_to_lds_b128 %0, %1, off"
               :: "v"(lds_addr), "v"(gaddr)
               : "memory");
#endif
}

__device__ __forceinline__ void wait_asynccnt0() {
#if defined(__has_builtin) && __has_builtin(__builtin_amdgcn_s_wait_asynccnt)
  __builtin_amdgcn_s_wait_asynccnt(0);
#else
  asm volatile("s_wait_asynccnt 0" ::: "memory");
#endif
  asm volatile("" ::: "memory");   // compiler fence before the LDS readback
}

// ---------------------------------------------------------------------------
// One block = ROWS_PER_BLOCK rows of one batch. 256 threads * float4 = 4KB
// row. Selection test is uniform per row -> SALU; bulk copy is NT b128
// streaming; interpolated rows stage substate[b,:] through LDS via the
// async data path, overlapped with the gather load.
// ---------------------------------------------------------------------------
__global__ __launch_bounds__(THREADS) void vbuf_scatter_kernel(
    const float* __restrict__ state,      // [B, O, D]
    const float* __restrict__ substate,   // [B, D]
    const int*   __restrict__ sel_index,  // [B, K]
    const float* __restrict__ sel_probs,  // [B, K]
    float* __restrict__ out)              // [B, O, D]
{
  __shared__ float lds_sub[D_DIM];        // 4 KB staging for substate row

  const int b   = blockIdx.y;
  const int tid = threadIdx.x;

  // Uniform selection metadata for this batch (scalar loads).
  int idx[K_SEL];
#pragma unroll
  for (int k = 0; k < K_SEL; ++k) idx[k] = sel_index[b * K_SEL + k];

#pragma unroll
  for (int j = 0; j < ROWS_PER_BLOCK; ++j) {
    const int o = blockIdx.x * ROWS_PER_BLOCK + j;
    const size_t row = ((size_t)b * O_BANKS + o) * (size_t)D_DIM;
    const v4f* src = (const v4f*)(state + row) + tid;
    v4f*       dst = (v4f*)(out + row) + tid;

    // Last duplicate wins (scatter-set order).
    int sel = -1;
#pragma unroll
    for (int k = 0; k < K_SEL; ++k)
      if (idx[k] == o) sel = k;

    if (sel < 0) {
      // Streaming copy: read-once / write-once -> non-temporal b128.
      v4f v = __builtin_nontemporal_load(src);
      __builtin_nontemporal_store(v, dst);
    } else {
      const float p = sel_probs[b * K_SEL + sel];
      const float q = 1.0f - p;

      // Kick off async global->LDS staging of the substate row, then issue
      // the gather load so both are in flight simultaneously.
      async_load_f4_to_lds(substate + (size_t)b * D_DIM + tid * 4,
                           &lds_sub[tid * 4]);
      v4f g = __builtin_nontemporal_load(src);
      wait_asynccnt0();                          // own-lane data: no barrier
      const v4f s = *(const v4f*)&lds_sub[tid * 4];

      v4f r;
      r.x = q * g.x + p * s.x;
      r.y = q * g.y + p * s.y;
      r.z = q * g.z + p * s.z;
      r.w = q * g.w + p * s.w;
      __builtin_nontemporal_store(r, dst);
    }
  }
}

extern "C" void kernel_launch(void* const* d_in, const int* in_sizes, int n_in,
                              void* d_out, int out_size, void* d_ws,
                              size_t ws_size, hipStream_t stream) {
  (void)n_in; (void)out_size; (void)d_ws; (void)ws_size;

  const float* state     = (const float*)d_in[0];
  const float* substate  = (const float*)d_in[1];
  const int*   sel_index = (const int*)d_in[2];   // int32 on device
  const float* sel_probs = (const float*)d_in[3];
  float*       out       = (float*)d_out;

  const int B = in_sizes[2] / K_SEL;              // sel_index is [B, K]

  dim3 grid(O_BANKS / ROWS_PER_BLOCK, B);         // (32, 2048)
  vbuf_scatter_kernel<<<grid, THREADS, 0, stream>>>(state, substate, sel_index,
                                                    sel_probs, out);
}